// RGCNModel_8495445312144
// MI455X (gfx1250) — compile-verified
//
#include <hip/hip_runtime.h>

typedef __attribute__((ext_vector_type(2))) float v2f;
typedef __attribute__((ext_vector_type(8))) float v8f;
typedef __attribute__((address_space(1))) int as1_int;
typedef __attribute__((address_space(3))) int as3_int;

#define R_REL 8

#if defined(__gfx1250__) && __has_builtin(__builtin_amdgcn_global_load_async_to_lds_b32)
#define USE_ASYNC_LDS 1
#else
#define USE_ASYNC_LDS 0
#endif

// ---------------------------------------------------------------------------
// GEMM: Y[n, o] = sum_f X[n, f] * W[f, o]   (W row-major FIN x FOUT, FIN=128)
//
// Block = 256 threads = 8 waves. W is staged once per block into LDS with a
// pair-interleaved layout using the gfx1250 async global->LDS path
// (GLOBAL_LOAD_ASYNC_TO_LDS_B32, ASYNCcnt): the async engine takes per-lane
// global AND per-lane LDS addresses, so the layout transform happens in the
// copy itself.
//
//   LDS layout:  ldsW2[k/2][n] = (W[k][n], W[k+1][n])  as float2,
//                row stride S2 = FOUT+16 float2s (S2 mod 32 == 16).
//
// Each wave computes a 16-row stripe across all FOUT columns with
// V_WMMA_F32_16X16X4_F32, K stepped by 4:
//   A fragment (16x4):  lanes 0-15 -> K = k,k+1 ; lanes 16-31 -> K = k+2,k+3
//   B fragment (4x16):  ONE ds_load_b64 -> (W[ka][n], W[ka+1][n]) in an
//                       even-aligned VGPR pair (no mov shuffles);
//                       n = (lane & 15) + 16*t
//   C/D (16x16):        VGPR j -> row m0 + j + 8*(lane>=16), col = lane & 15
// Bank use per b64 load: lanes 0-15 span 32 bank slots, lanes 16-31 are
// offset by 2*S2 mod 64 == 32 -> all 64 banks, conflict-free.
// ---------------------------------------------------------------------------
template <int NT>  // NT = FOUT / 16
__launch_bounds__(256)
__global__ void gemm_wmma_lds(const float* __restrict__ X,
                              const float* __restrict__ W,
                              float* __restrict__ Y,
                              int nrows) {
  constexpr int FIN = 128;
  constexpr int FOUT = NT * 16;
  constexpr int S2 = FOUT + 16;  // float2 row stride (bank-conflict padding)
  __shared__ v2f ldsW[(FIN / 2) * S2];

  const int tid = (int)threadIdx.x;

  // ---- stage W (FIN x FOUT) into pair-interleaved LDS ----
  {
    float* lds_f = (float*)ldsW;
    for (int id = tid; id < FIN * FOUT; id += 256) {
      const int k = id / FOUT;       // row (consecutive tids walk n: coalesced)
      const int n = id % FOUT;
      const int fidx = ((k >> 1) * S2 + n) * 2 + (k & 1);
#if USE_ASYNC_LDS
      __builtin_amdgcn_global_load_async_to_lds_b32(
          (as1_int*)(W + id), (as3_int*)(lds_f + fidx), /*offset=*/0, /*cpol=*/0);
#else
      lds_f[fidx] = W[id];
#endif
    }
  }
#if USE_ASYNC_LDS
#if __has_builtin(__builtin_amdgcn_s_wait_asynccnt)
  __builtin_amdgcn_s_wait_asynccnt(0);
#else
  asm volatile("s_wait_asynccnt 0x0" ::: "memory");
#endif
#endif
  __syncthreads();

  const int wave = tid >> 5;
  const int lane = tid & 31;
  const int m0 = ((int)blockIdx.x * 8 + wave) * 16;
  if (m0 >= nrows) return;  // no barriers past this point

  const int half = lane >> 4;  // 0: K pair (k,k+1), 1: K pair (k+2,k+3)
  const int lm   = lane & 15;

  v8f acc[NT] = {};
  const float* __restrict__ xrow = X + (size_t)(m0 + lm) * FIN;

  for (int k = 0; k < FIN; k += 4) {
    const int ka = k + half * 2;
    v2f a;
    a.x = xrow[ka];
    a.y = xrow[ka + 1];
    const v2f* __restrict__ lrow = ldsW + ((k >> 1) + half) * S2 + lm;
#pragma unroll
    for (int t = 0; t < NT; ++t) {
      const v2f b = lrow[t * 16];  // single ds_load_b64, even-aligned pair
      acc[t] = __builtin_amdgcn_wmma_f32_16x16x4_f32(
          false, a, false, b, (short)0, acc[t], false, false);
    }
  }

#pragma unroll
  for (int t = 0; t < NT; ++t) {
#pragma unroll
    for (int j = 0; j < 8; ++j) {
      const int row = m0 + j + half * 8;
      const int col = t * 16 + lm;
      Y[(size_t)row * FOUT + col] = acc[t][j];
    }
  }
}

// ---------------------------------------------------------------------------
// Per-(dst, relation) edge counts (mean normalization denominators).
// ---------------------------------------------------------------------------
__global__ void count_edges(const int* __restrict__ dst,
                            const int* __restrict__ et,
                            int* __restrict__ counts, int E) {
  const int e = (int)(blockIdx.x * blockDim.x + threadIdx.x);
  if (e >= E) return;
  atomicAdd(&counts[(size_t)dst[e] * R_REL + et[e]], 1);
}

// ---------------------------------------------------------------------------
// Scatter for one relation: acc[dst] += XW_r[src] / count[dst, r].
// Each thread handles one edge and 4 consecutive features.
// fshift = log2(fout/4) : 5 for F=128, 4 for F=64.
// ---------------------------------------------------------------------------
__launch_bounds__(256)
__global__ void scatter_rel(const float* __restrict__ xw,
                            const int* __restrict__ src,
                            const int* __restrict__ dst,
                            const int* __restrict__ et,
                            const int* __restrict__ counts,
                            float* __restrict__ acc,
                            int E, int fout, int rel, int fshift) {
  const long long tid = (long long)blockIdx.x * blockDim.x + threadIdx.x;
  const int e = (int)(tid >> fshift);
  if (e >= E) return;
  if (et[e] != rel) return;
  const int f = ((int)tid & ((1 << fshift) - 1)) << 2;

  const int s = src[e];
  const int d = dst[e];
  const float inv = 1.0f / (float)counts[(size_t)d * R_REL + rel];

  const float4 v = *(const float4*)(xw + (size_t)s * fout + f);
  float* __restrict__ o = acc + (size_t)d * fout + f;
  atomicAdd(o + 0, v.x * inv);
  atomicAdd(o + 1, v.y * inv);
  atomicAdd(o + 2, v.z * inv);
  atomicAdd(o + 3, v.w * inv);
}

// ---------------------------------------------------------------------------
// acc[i] = act(acc[i] + xroot[i] + bias[i % fout]) ; act = relu or sigmoid.
// ---------------------------------------------------------------------------
__global__ void finalize_act(float* __restrict__ acc,
                             const float* __restrict__ xroot,
                             const float* __restrict__ bias,
                             int total, int fout, int sigmoid) {
  const int i = (int)(blockIdx.x * blockDim.x + threadIdx.x);
  if (i >= total) return;
  float v = acc[i] + xroot[i] + bias[i % fout];
  acc[i] = sigmoid ? (1.0f / (1.0f + __expf(-v))) : fmaxf(v, 0.0f);
}

// ---------------------------------------------------------------------------
// Host orchestration.
// ---------------------------------------------------------------------------
extern "C" void kernel_launch(void* const* d_in, const int* in_sizes, int n_in,
                              void* d_out, int out_size, void* d_ws, size_t ws_size,
                              hipStream_t stream) {
  const float* x     = (const float*)d_in[0];
  const int*   eidx  = (const int*)d_in[1];
  const int*   etype = (const int*)d_in[2];
  const float* W1    = (const float*)d_in[3];
  const float* root1 = (const float*)d_in[4];
  const float* b1    = (const float*)d_in[5];
  const float* W2    = (const float*)d_in[6];
  const float* root2 = (const float*)d_in[7];
  const float* b2    = (const float*)d_in[8];
  float* out = (float*)d_out;

  const int N = 50000, E = 500000, FIN = 128, FHID = 128, FOUT = 64;
  const int* src = eidx;
  const int* dst = eidx + E;

  // Workspace layout (bytes):
  //   counts : N*R int   =  1,600,000
  //   xw     : N*128 f32 = 25,600,000  (per-relation transform, L2-resident)
  //   xroot  : N*128 f32 = 25,600,000  (root transform; N*64 reuse in layer 2)
  //   hacc   : N*128 f32 = 25,600,000  (layer-1 accumulator -> hidden h)
  char* ws = (char*)d_ws;
  int*   counts = (int*)ws;
  float* xw     = (float*)(ws + 1600000);
  float* xroot  = (float*)(ws + 1600000 + 25600000);
  float* hacc   = (float*)(ws + 1600000 + 2 * (size_t)25600000);

  const int gemmBlocks = (N / 16 + 7) / 8;  // 8 waves/block, 16 rows/wave

  // ---- degree counts (shared by both layers) ----
  (void)hipMemsetAsync(counts, 0, (size_t)N * R_REL * sizeof(int), stream);
  count_edges<<<(E + 255) / 256, 256, 0, stream>>>(dst, etype, counts, E);

  // ---- layer 1 ----
  gemm_wmma_lds<8><<<gemmBlocks, 256, 0, stream>>>(x, root1, xroot, N);
  (void)hipMemsetAsync(hacc, 0, (size_t)N * FHID * sizeof(float), stream);
  for (int r = 0; r < R_REL; ++r) {
    gemm_wmma_lds<8><<<gemmBlocks, 256, 0, stream>>>(
        x, W1 + (size_t)r * FIN * FHID, xw, N);
    const long long th = (long long)E * (FHID / 4);
    scatter_rel<<<(int)((th + 255) / 256), 256, 0, stream>>>(
        xw, src, dst, etype, counts, hacc, E, FHID, r, 5);
  }
  finalize_act<<<(N * FHID + 255) / 256, 256, 0, stream>>>(
      hacc, xroot, b1, N * FHID, FHID, 0);  // hacc now holds h = relu(...)

  // ---- layer 2 ----
  gemm_wmma_lds<4><<<gemmBlocks, 256, 0, stream>>>(hacc, root2, xroot, N);
  (void)hipMemsetAsync(out, 0, (size_t)N * FOUT * sizeof(float), stream);
  for (int r = 0; r < R_REL; ++r) {
    gemm_wmma_lds<4><<<gemmBlocks, 256, 0, stream>>>(
        hacc, W2 + (size_t)r * FHID * FOUT, xw, N);
    const long long th = (long long)E * (FOUT / 4);
    scatter_rel<<<(int)((th + 255) / 256), 256, 0, stream>>>(
        xw, src, dst, etype, counts, out, E, FOUT, r, 4);
  }
  finalize_act<<<(N * FOUT + 255) / 256, 256, 0, stream>>>(
      out, xroot, b2, N * FOUT, FOUT, 1);
}